// simpleGIN_55207509623273
// MI455X (gfx1250) — compile-verified
//
#include <hip/hip_runtime.h>
#include <hip/hip_bf16.h>

#define N_NODES  100000
#define N_EDGES  3200000
#define N_GRAPHS 1024
#define DIM      128
#define BN_EPS   1e-5f
#define NPAD     132   // 16-row LDS tile row stride (floats), spreads banks
#define NUM_CHUNKS ((N_NODES + 255) / 256)   // 391

typedef __attribute__((ext_vector_type(2))) float v2f;
typedef __attribute__((ext_vector_type(8))) float v8f;

__device__ __forceinline__ void atomAddF(float* p, float v) {
    __hip_atomic_fetch_add(p, v, __ATOMIC_RELAXED, __HIP_MEMORY_SCOPE_AGENT);
}

// ===========================================================================
// CSR construction (once per call): rowptr over dst, edge sources sorted by dst
// ===========================================================================
__global__ __launch_bounds__(256) void csr_degree(
    const int* __restrict__ dst, int* __restrict__ deg)
{
    const int e = blockIdx.x * blockDim.x + threadIdx.x;
    if (e >= N_EDGES) return;
    atomicAdd(&deg[dst[e]], 1);
}

__global__ __launch_bounds__(256) void csr_chunk_sum(
    const int* __restrict__ deg, int* __restrict__ chunkSum)
{
    __shared__ int sh[256];
    const int t = threadIdx.x;
    const int i = blockIdx.x * 256 + t;
    sh[t] = (i < N_NODES) ? deg[i] : 0;
    __syncthreads();
    #pragma unroll
    for (int off = 128; off > 0; off >>= 1) {
        if (t < off) sh[t] += sh[t + off];
        __syncthreads();
    }
    if (t == 0) chunkSum[blockIdx.x] = sh[0];
}

__global__ void csr_scan_chunks(const int* __restrict__ chunkSum,
                                int* __restrict__ chunkOff, int* __restrict__ rowptr)
{
    if (threadIdx.x == 0) {
        int run = 0;
        for (int b = 0; b < NUM_CHUNKS; ++b) { chunkOff[b] = run; run += chunkSum[b]; }
        rowptr[N_NODES] = run;
    }
}

__global__ __launch_bounds__(256) void csr_scan_within(
    const int* __restrict__ deg, const int* __restrict__ chunkOff,
    int* __restrict__ rowptr, int* __restrict__ cursor)
{
    __shared__ int sh[256];
    const int t = threadIdx.x;
    const int i = blockIdx.x * 256 + t;
    const int v = (i < N_NODES) ? deg[i] : 0;
    sh[t] = v;
    __syncthreads();
    #pragma unroll
    for (int off = 1; off < 256; off <<= 1) {   // Hillis-Steele inclusive scan
        const int x = (t >= off) ? sh[t - off] : 0;
        __syncthreads();
        sh[t] += x;
        __syncthreads();
    }
    if (i < N_NODES) {
        const int ex = chunkOff[blockIdx.x] + sh[t] - v;  // exclusive
        rowptr[i] = ex;
        cursor[i] = ex;
    }
}

__global__ __launch_bounds__(256) void csr_scatter(
    const int* __restrict__ src, const int* __restrict__ dst,
    int* __restrict__ cursor, int* __restrict__ esorted)
{
    const int e = blockIdx.x * blockDim.x + threadIdx.x;
    if (e >= N_EDGES) return;
    const int p = atomicAdd(&cursor[dst[e]], 1);
    esorted[p] = src[e];
}

// per-graph [start,end) bounds from the sorted batch vector (once per call)
__global__ void graph_bounds_init(int* __restrict__ gstart, int* __restrict__ gend)
{
    const int g = blockIdx.x * blockDim.x + threadIdx.x;
    if (g < N_GRAPHS) { gstart[g] = N_NODES; gend[g] = 0; }
}

__global__ __launch_bounds__(256) void graph_bounds(
    const int* __restrict__ batch, int* __restrict__ gstart, int* __restrict__ gend)
{
    const int i = blockIdx.x * blockDim.x + threadIdx.x;
    if (i >= N_NODES) return;
    const int g = batch[i];
    atomicMin(&gstart[g], i);
    atomicMax(&gend[g], i + 1);
}

// ===========================================================================
// Gather-based GIN aggregation: U[n] = h[n] + sum_{s in N_in(n)} h[s].
// One wave per node; each lane carries a float4 (32*16B = full 512B row).
// Pure coalesced reads, zero atomics. 2-way unrolled for outstanding loads.
// ===========================================================================
__global__ __launch_bounds__(256) void gin_aggregate(
    const float* __restrict__ h, const int* __restrict__ rowptr,
    const int* __restrict__ esorted, float* __restrict__ U)
{
    const int lane = threadIdx.x & 31;
    const int node = (int)((blockIdx.x * (unsigned)blockDim.x + threadIdx.x) >> 5);
    if (node >= N_NODES) return;
    const int beg = rowptr[node];
    const int end = rowptr[node + 1];
    float4 acc = ((const float4*)(h + (size_t)node * DIM))[lane];
    int e = beg;
    for (; e + 1 < end; e += 2) {
        const int s0 = esorted[e];
        const int s1 = esorted[e + 1];
        const float4 v0 = ((const float4*)(h + (size_t)s0 * DIM))[lane];
        const float4 v1 = ((const float4*)(h + (size_t)s1 * DIM))[lane];
        acc.x += v0.x + v1.x; acc.y += v0.y + v1.y;
        acc.z += v0.z + v1.z; acc.w += v0.w + v1.w;
    }
    if (e < end) {
        const int s = esorted[e];
        const float4 v = ((const float4*)(h + (size_t)s * DIM))[lane];
        acc.x += v.x; acc.y += v.y; acc.z += v.z; acc.w += v.w;
    }
    ((float4*)(U + (size_t)node * DIM))[lane] = acc;
}

// ===========================================================================
// Stage 1: Y = U @ W + b  and column sum / sumsq for BatchNorm.
// Block = 8 waves; block covers rows [16*bid, +16), wave w covers cols [16w, +16)
// via V_WMMA_F32_16X16X4_F32 (32 K-steps of K=4).
// ===========================================================================
__global__ __launch_bounds__(256) void gin_gemm1(
    const float* __restrict__ U,
    const float* __restrict__ W, const float* __restrict__ bias,
    float* __restrict__ Y, float* __restrict__ stat /*[0..127]=sum [128..255]=sumsq*/)
{
    __shared__ float tile[16 * NPAD];
    const int tid = threadIdx.x;
    const int rowBase = blockIdx.x * 16;

    #pragma unroll
    for (int i = 0; i < 2; ++i) {
        const int e  = tid + i * 256;    // float4 index 0..511
        const int r  = e >> 5;
        const int c4 = e & 31;
        const float4 a = *(const float4*)(U + (size_t)(rowBase + r) * DIM + c4 * 4);
        float* t = &tile[r * NPAD + c4 * 4];
        t[0] = a.x; t[1] = a.y; t[2] = a.z; t[3] = a.w;
    }
    __syncthreads();

    const int lane = tid & 31;
    const int wave = tid >> 5;
    const int half = lane >> 4;          // 0: K 0..1 / rows 0..7 ; 1: K 2..3 / rows 8..15
    const int ml   = lane & 15;
    const int col  = wave * 16 + ml;

    v8f acc;
    const float bv = bias[col];
    #pragma unroll
    for (int j = 0; j < 8; ++j) acc[j] = bv;

    #pragma unroll
    for (int kk = 0; kk < 32; ++kk) {
        const int k0 = kk * 4 + half * 2;
        v2f a, b;
        a.x = tile[ml * NPAD + k0];
        a.y = tile[ml * NPAD + k0 + 1];
        b.x = W[(size_t)k0 * DIM + col];
        b.y = W[(size_t)(k0 + 1) * DIM + col];
        acc = __builtin_amdgcn_wmma_f32_16x16x4_f32(
            false, a, false, b, (short)0, acc, false, false);
    }

    float s = 0.f, q = 0.f;
    #pragma unroll
    for (int j = 0; j < 8; ++j) {
        const int row = rowBase + half * 8 + j;
        const float v = acc[j];
        Y[(size_t)row * DIM + col] = v;
        s += v; q += v * v;
    }
    s += __shfl_xor(s, 16, 32);          // lanes l and l+16 share a column
    q += __shfl_xor(q, 16, 32);
    if (half == 0) {
        atomAddF(&stat[col], s);
        atomAddF(&stat[DIM + col], q);
    }
}

__global__ void gin_bn_finalize(const float* __restrict__ stat,
                                float* __restrict__ meanv, float* __restrict__ invv)
{
    const int c = threadIdx.x;   // 128 threads
    const float invN = 1.0f / (float)N_NODES;
    const float m = stat[c] * invN;
    const float v = stat[DIM + c] * invN - m * m;
    meanv[c] = m;
    invv[c]  = rsqrtf(v + BN_EPS);
}

// ===========================================================================
// Stage 2: Hout = relu( relu(bn(Y)) @ W2 + b2 ); BN+ReLU fused into LDS fill.
// ===========================================================================
__global__ __launch_bounds__(256) void gin_gemm2(
    const float* __restrict__ Y,
    const float* __restrict__ gamma, const float* __restrict__ beta,
    const float* __restrict__ meanv, const float* __restrict__ invv,
    const float* __restrict__ W, const float* __restrict__ bias,
    float* __restrict__ Hout)
{
    __shared__ float tile[16 * NPAD];
    const int tid = threadIdx.x;
    const int rowBase = blockIdx.x * 16;

    #pragma unroll
    for (int i = 0; i < 2; ++i) {
        const int e  = tid + i * 256;
        const int r  = e >> 5;
        const int c4 = e & 31;
        const float4 y4 = *(const float4*)(Y + (size_t)(rowBase + r) * DIM + c4 * 4);
        const float4 g4 = *(const float4*)(gamma + c4 * 4);
        const float4 b4 = *(const float4*)(beta  + c4 * 4);
        const float4 m4 = *(const float4*)(meanv + c4 * 4);
        const float4 i4 = *(const float4*)(invv  + c4 * 4);
        float* t = &tile[r * NPAD + c4 * 4];
        t[0] = fmaxf(g4.x * (y4.x - m4.x) * i4.x + b4.x, 0.f);
        t[1] = fmaxf(g4.y * (y4.y - m4.y) * i4.y + b4.y, 0.f);
        t[2] = fmaxf(g4.z * (y4.z - m4.z) * i4.z + b4.z, 0.f);
        t[3] = fmaxf(g4.w * (y4.w - m4.w) * i4.w + b4.w, 0.f);
    }
    __syncthreads();

    const int lane = tid & 31;
    const int wave = tid >> 5;
    const int half = lane >> 4;
    const int ml   = lane & 15;
    const int col  = wave * 16 + ml;

    v8f acc;
    const float bv = bias[col];
    #pragma unroll
    for (int j = 0; j < 8; ++j) acc[j] = bv;

    #pragma unroll
    for (int kk = 0; kk < 32; ++kk) {
        const int k0 = kk * 4 + half * 2;
        v2f a, b;
        a.x = tile[ml * NPAD + k0];
        a.y = tile[ml * NPAD + k0 + 1];
        b.x = W[(size_t)k0 * DIM + col];
        b.y = W[(size_t)(k0 + 1) * DIM + col];
        acc = __builtin_amdgcn_wmma_f32_16x16x4_f32(
            false, a, false, b, (short)0, acc, false, false);
    }

    #pragma unroll
    for (int j = 0; j < 8; ++j) {
        const int row = rowBase + half * 8 + j;
        Hout[(size_t)row * DIM + col] = fmaxf(acc[j], 0.f);
    }
}

// ===========================================================================
// Sorted-batch pooling: one wave per graph streams its contiguous node rows.
// No atomics; empty graphs write zeros (start=N, end=0 -> loop skipped).
// ===========================================================================
__global__ __launch_bounds__(256) void gin_pool_sorted(
    const float* __restrict__ H, const int* __restrict__ gstart,
    const int* __restrict__ gend, float* __restrict__ pooled, int layerOff)
{
    const int lane = threadIdx.x & 31;
    const int g = (int)((blockIdx.x * (unsigned)blockDim.x + threadIdx.x) >> 5);
    if (g >= N_GRAPHS) return;
    const int b = gstart[g];
    const int e = gend[g];
    float4 acc = make_float4(0.f, 0.f, 0.f, 0.f);
    for (int i = b; i < e; ++i) {
        const float4 v = ((const float4*)(H + (size_t)i * DIM))[lane];
        acc.x += v.x; acc.y += v.y; acc.z += v.z; acc.w += v.w;
    }
    *(float4*)(pooled + (size_t)g * (4 * DIM) + layerOff + lane * 4) = acc;
}

// ===========================================================================
// Small classifier GEMM: out[M,Nc] = (relu?)(A[M,K] @ W[K,Nc] + b)
// ===========================================================================
__global__ __launch_bounds__(256) void gin_linear(
    const float* __restrict__ A, const float* __restrict__ W,
    const float* __restrict__ b, float* __restrict__ out,
    int M, int K, int Nc, int doRelu)
{
    const int idx = blockIdx.x * blockDim.x + threadIdx.x;
    if (idx >= M * Nc) return;
    const int m = idx / Nc;
    const int n = idx % Nc;
    float acc = b[n];
    for (int k = 0; k < K; ++k)
        acc = fmaf(A[(size_t)m * K + k], W[(size_t)k * Nc + n], acc);
    out[idx] = doRelu ? fmaxf(acc, 0.f) : acc;
}

// ===========================================================================
extern "C" void kernel_launch(void* const* d_in, const int* in_sizes, int n_in,
                              void* d_out, int out_size, void* d_ws, size_t ws_size,
                              hipStream_t stream)
{
    (void)in_sizes; (void)n_in; (void)out_size; (void)ws_size;

    const float* x     = (const float*)d_in[0];
    const int*   ei    = (const int*)  d_in[1];
    const int*   batch = (const int*)  d_in[2];
    const float* w1  = (const float*)d_in[3];
    const float* b1  = (const float*)d_in[4];
    const float* g1  = (const float*)d_in[5];
    const float* be1 = (const float*)d_in[6];
    const float* w2  = (const float*)d_in[7];
    const float* b2  = (const float*)d_in[8];
    const float* lw1  = (const float*)d_in[9];
    const float* lb1  = (const float*)d_in[10];
    const float* lg1  = (const float*)d_in[11];
    const float* lbe1 = (const float*)d_in[12];
    const float* lw2  = (const float*)d_in[13];
    const float* lb2  = (const float*)d_in[14];
    const float* cw1 = (const float*)d_in[15];
    const float* cb1 = (const float*)d_in[16];
    const float* cw2 = (const float*)d_in[17];
    const float* cb2 = (const float*)d_in[18];
    const float* cw3 = (const float*)d_in[19];
    const float* cb3 = (const float*)d_in[20];

    const int* srcI = ei;
    const int* dstI = ei + N_EDGES;

    // ---- workspace layout ----
    const size_t NH = (size_t)N_NODES * DIM;    // 12.8M floats per node buffer
    float* ws = (float*)d_ws;
    float* Hb     = ws;                          // h buffer (GEMM2 output)
    float* Ub     = Hb + NH;                     // aggregated MLP input
    float* Yb     = Ub + NH;                     // post-linear1
    float* stat   = Yb + NH;                     // 256 (sum, sumsq)
    float* meanv  = stat + 256;                  // 128
    float* invv   = meanv + DIM;                 // 128
    float* pooled = invv + DIM;                  // 1024*512
    float* t1     = pooled + (size_t)N_GRAPHS * 4 * DIM;   // 1024*128
    float* t2     = t1 + (size_t)N_GRAPHS * DIM;           // 1024*64
    int* iws      = (int*)(t2 + (size_t)N_GRAPHS * (DIM / 2));
    int* deg      = iws;                         // N_NODES
    int* rowptr   = deg + N_NODES;               // N_NODES + 1
    int* cursor   = rowptr + (N_NODES + 1);      // N_NODES
    int* esorted  = cursor + N_NODES;            // N_EDGES
    int* chunkSum = esorted + N_EDGES;           // NUM_CHUNKS
    int* chunkOff = chunkSum + NUM_CHUNKS;       // NUM_CHUNKS
    int* gstart   = chunkOff + NUM_CHUNKS;       // N_GRAPHS
    int* gend     = gstart + N_GRAPHS;           // N_GRAPHS

    const int rowTiles   = N_NODES / 16;                 // 6250 exact
    const int edgeBlocks = (N_EDGES + 255) / 256;        // 12500
    const int nodeBlocks = (N_NODES + 255) / 256;        // 391
    const int nodeWaveBl = (N_NODES * 32 + 255) / 256;   // 12500
    const int grphWaveBl = (N_GRAPHS * 32 + 255) / 256;  // 128

    // ---- one-time CSR + graph-bounds construction ----
    hipMemsetAsync(deg, 0, N_NODES * sizeof(int), stream);
    csr_degree<<<edgeBlocks, 256, 0, stream>>>(dstI, deg);
    csr_chunk_sum<<<NUM_CHUNKS, 256, 0, stream>>>(deg, chunkSum);
    csr_scan_chunks<<<1, 1, 0, stream>>>(chunkSum, chunkOff, rowptr);
    csr_scan_within<<<NUM_CHUNKS, 256, 0, stream>>>(deg, chunkOff, rowptr, cursor);
    csr_scatter<<<edgeBlocks, 256, 0, stream>>>(srcI, dstI, cursor, esorted);
    graph_bounds_init<<<(N_GRAPHS + 255) / 256, 256, 0, stream>>>(gstart, gend);
    graph_bounds<<<nodeBlocks, 256, 0, stream>>>(batch, gstart, gend);

    // ---- 4 GIN layers ----
    const float* hcur = x;
    for (int layer = 0; layer < 4; ++layer) {
        const float* W1; const float* B1; const float* G; const float* BE;
        const float* W2; const float* B2;
        if (layer == 0) { W1 = w1; B1 = b1; G = g1; BE = be1; W2 = w2; B2 = b2; }
        else {
            const int j = layer - 1;
            W1 = lw1 + (size_t)j * DIM * DIM;  B1 = lb1  + (size_t)j * DIM;
            G  = lg1 + (size_t)j * DIM;        BE = lbe1 + (size_t)j * DIM;
            W2 = lw2 + (size_t)j * DIM;        B2 = lb2  + (size_t)j * DIM;
        }
        if (layer > 0) W2 = lw2 + (size_t)(layer - 1) * DIM * DIM;  // full matrix stride

        hipMemsetAsync(stat, 0, 256 * sizeof(float), stream);
        gin_aggregate<<<nodeWaveBl, 256, 0, stream>>>(hcur, rowptr, esorted, Ub);
        gin_gemm1<<<rowTiles, 256, 0, stream>>>(Ub, W1, B1, Yb, stat);
        gin_bn_finalize<<<1, DIM, 0, stream>>>(stat, meanv, invv);
        gin_gemm2<<<rowTiles, 256, 0, stream>>>(Yb, G, BE, meanv, invv, W2, B2, Hb);
        gin_pool_sorted<<<grphWaveBl, 256, 0, stream>>>(Hb, gstart, gend, pooled, layer * DIM);
        hcur = Hb;
    }

    // ---- classifier: [1024,512] -> relu[1024,128] -> relu[1024,64] -> [1024,2]
    gin_linear<<<(N_GRAPHS * DIM + 255) / 256, 256, 0, stream>>>(
        pooled, cw1, cb1, t1, N_GRAPHS, 4 * DIM, DIM, 1);
    gin_linear<<<(N_GRAPHS * (DIM / 2) + 255) / 256, 256, 0, stream>>>(
        t1, cw2, cb2, t2, N_GRAPHS, DIM, DIM / 2, 1);
    gin_linear<<<(N_GRAPHS * 2 + 255) / 256, 256, 0, stream>>>(
        t2, cw3, cb3, (float*)d_out, N_GRAPHS, DIM / 2, 2, 0);
}